// Cst2Val_Layer_38190849196760
// MI455X (gfx1250) — compile-verified
//
#include <hip/hip_runtime.h>

#define HDIM 128

typedef __attribute__((ext_vector_type(2))) float v2f;
typedef __attribute__((ext_vector_type(8))) float v8f;

__device__ __forceinline__ v8f wmma_f32(v2f a, v2f b, v8f c) {
  // V_WMMA_F32_16X16X4_F32: D = A(16x4) * B(4x16) + C(16x16), fp32 exact
  return __builtin_amdgcn_wmma_f32_16x16x4_f32(
      /*neg_a=*/false, a, /*neg_b=*/false, b,
      /*c_mod=*/(short)0, c, /*reuse_a=*/false, /*reuse_b=*/false);
}

// ---------------------------------------------------------------------------
// Async memory->LDS copy (CDNA5): per-lane 16B, tracked by ASYNCcnt.
// lds_off is the workgroup-relative LDS byte offset (low 32 bits of a generic
// pointer to __shared__, per the aperture mapping in ISA 10.2).
// ---------------------------------------------------------------------------
__device__ __forceinline__ void async_copy_b128(unsigned lds_off, const void* gaddr) {
  asm volatile("global_load_async_to_lds_b128 %0, %1, off"
               :: "v"(lds_off), "v"(gaddr)
               : "memory");
}
__device__ __forceinline__ void wait_async0() {
  asm volatile("s_wait_asynccnt 0" ::: "memory");
}
__device__ __forceinline__ unsigned lds_offset(const void* p) {
  return (unsigned)(unsigned long long)p;   // low 32 bits = LDS offset
}

// Stage a 16x128 f32 tile (8 KB) from global into LDS with async copies.
// One wave: 16 async b128 ops, 512 B each.
__device__ __forceinline__ void stage_tile_async(float* sDst, const float* gSrc, int lane) {
  const unsigned base = lds_offset(sDst);
  for (int b = lane * 16; b < 16 * 128 * 4; b += 32 * 16)
    async_copy_b128(base + b, (const char*)gSrc + b);
}

// One 16x16 output tile of  C = A(16xK, LDS, ld=K) * B(KxN, global row-major).
// lo = lane&15, hi = lane>>4.  C layout: acc[r] -> row (r + 8*hi), col (ntile*16 + lo).
__device__ __forceinline__ v8f gemm_tile(const float* __restrict__ sA,
                                         const float* __restrict__ Bg,
                                         int K, int N, int ntile, int lo, int hi) {
  v8f acc = {};
  const int col = ntile * 16 + lo;
  for (int k = 0; k < K; k += 4) {
    const int kb = k + 2 * hi;
    v2f a = *(const v2f*)(sA + lo * K + kb);              // A[m][kb], A[m][kb+1]
    const float* bp = Bg + (size_t)kb * N + col;
    v2f b = { bp[0], bp[N] };                             // B[kb][col], B[kb+1][col]
    acc = wmma_f32(a, b, acc);
  }
  return acc;
}

// ---------------------------------------------------------------------------
// Kernel 1: m = LayerNorm_512( ReLU(r_cst @ W1 + b1) @ W2 ) * g1 + bt1
// One wave (32 threads) per 16-row tile.
// ---------------------------------------------------------------------------
__global__ __launch_bounds__(32)
void cst_send_kernel(const float* __restrict__ r_cst,
                     const float* __restrict__ W1, const float* __restrict__ b1,
                     const float* __restrict__ W2,
                     const float* __restrict__ g1, const float* __restrict__ bt1,
                     float* __restrict__ m_out) {
  __shared__ float sA[16 * 128];   // staged r_cst tile
  __shared__ float sH[16 * 128];   // relu(A@W1+b1)
  __shared__ float sO[16 * 512];   // raw GEMM2 output
  __shared__ float sMu[16], sRs[16];

  const int lane = threadIdx.x;
  const int lo = lane & 15, hi = lane >> 4;
  const int tile = blockIdx.x;

  // stage 16x128 input tile: async global->LDS (CDNA5 path)
  stage_tile_async(sA, r_cst + (size_t)tile * 16 * 128, lane);
  wait_async0();
  __syncthreads();

  // GEMM1 + bias + ReLU  (N = 128 -> 8 tiles)
  for (int nt = 0; nt < 8; ++nt) {
    v8f acc = gemm_tile(sA, W1, 128, 128, nt, lo, hi);
    const float bias = b1[nt * 16 + lo];
    for (int r = 0; r < 8; ++r) {
      float v = acc[r] + bias;
      v = v > 0.0f ? v : 0.0f;
      sH[(r + 8 * hi) * 128 + nt * 16 + lo] = v;
    }
  }
  __syncthreads();

  // GEMM2 (N = 512 -> 32 tiles) with fused sum / sum-of-squares
  float rs[8], rq[8];
  for (int r = 0; r < 8; ++r) { rs[r] = 0.0f; rq[r] = 0.0f; }
  for (int nt = 0; nt < 32; ++nt) {
    v8f acc = gemm_tile(sH, W2, 128, 512, nt, lo, hi);
    for (int r = 0; r < 8; ++r) {
      const float v = acc[r];
      sO[(r + 8 * hi) * 512 + nt * 16 + lo] = v;
      rs[r] += v;
      rq[r] += v * v;
    }
  }
  // reduce across the 16 lanes that share a row (lanes 0-15 own rows 0-7, 16-31 own 8-15)
  for (int r = 0; r < 8; ++r) {
    float s = rs[r], q = rq[r];
    for (int off = 8; off > 0; off >>= 1) {
      s += __shfl_xor(s, off, 16);
      q += __shfl_xor(q, off, 16);
    }
    if (lo == 0) {
      const int row = r + 8 * hi;
      const float mu = s * (1.0f / 512.0f);
      const float var = q * (1.0f / 512.0f) - mu * mu;
      sMu[row] = mu;
      sRs[row] = rsqrtf(var + 1e-5f);
    }
  }
  __syncthreads();

  // normalize + affine, write out
  float* dst = m_out + (size_t)tile * 16 * 512;
  for (int i = lane; i < 16 * 512 / 4; i += 32) {
    const int row = i >> 7;            // 128 float4 per row
    const int c4 = (i & 127) * 4;
    float4 v = *(float4*)(sO + row * 512 + c4);
    const float mu = sMu[row], rstd = sRs[row];
    const float4 g = *(const float4*)(g1 + c4);
    const float4 bb = *(const float4*)(bt1 + c4);
    v.x = (v.x - mu) * rstd * g.x + bb.x;
    v.y = (v.y - mu) * rstd * g.y + bb.y;
    v.z = (v.z - mu) * rstd * g.z + bb.z;
    v.w = (v.w - mu) * rstd * g.w + bb.w;
    *(float4*)(dst + row * 512 + c4) = v;
  }
}

// ---------------------------------------------------------------------------
// Kernel 2: r_val[in_idx] += m[4*cst + 2*LE + PE]   (segment_sum via f32 atomics)
// 256 threads handle 64 edges; edge indices staged once through LDS.
// ---------------------------------------------------------------------------
__global__ __launch_bounds__(256)
void edge_scatter_kernel(const float* __restrict__ m,
                         const int* __restrict__ e_out, const int* __restrict__ e_in,
                         const int* __restrict__ LE, const int* __restrict__ PE,
                         float* __restrict__ r_val, int E) {
  __shared__ int sOI[64];
  __shared__ int sII[64];
  const int t = threadIdx.x;
  const long long ebase = (long long)blockIdx.x * 64;
  if (t < 64) {
    const long long e = ebase + t;
    int oi = -1, ii = 0;
    if (e < E) {
      oi = 4 * e_out[e] + 2 * LE[e] + PE[e];
      ii = e_in[e];
    }
    sOI[t] = oi;
    sII[t] = ii;
  }
  __syncthreads();
  for (int j = 0; j < 8; ++j) {
    const int chunk = t + 256 * j;          // 0..2047 = 64 edges * 32 float4
    const int el = chunk >> 5;
    const int c4 = (chunk & 31) * 4;
    const int oi = sOI[el];
    if (oi < 0) continue;
    const int ii = sII[el];
    const float4 v = *(const float4*)(m + (size_t)oi * HDIM + c4);
    float* d = r_val + (size_t)ii * HDIM + c4;
    atomicAdd(d + 0, v.x);
    atomicAdd(d + 1, v.y);
    atomicAdd(d + 2, v.z);
    atomicAdd(d + 3, v.w);
  }
}

// ---------------------------------------------------------------------------
// Kernel 3: out = LayerNorm_128( ReLU((x+r) @ W3 + b3) @ W4 ) * g2 + bt2 + x
// r_val lives in d_out; output overwrites d_out (tile-local in-place, safe).
// ---------------------------------------------------------------------------
__global__ __launch_bounds__(32)
void val_rec_kernel(const float* __restrict__ x_val, const float* __restrict__ r_val,
                    const float* __restrict__ W3, const float* __restrict__ b3,
                    const float* __restrict__ W4,
                    const float* __restrict__ g2, const float* __restrict__ bt2,
                    float* __restrict__ out) {
  __shared__ float sX[16 * 128];   // x tile, becomes h = x + r
  __shared__ float sR[16 * 128];   // r tile
  __shared__ float sT[16 * 128];   // relu(h@W3+b3)
  __shared__ float sO[16 * 128];   // t@W4
  __shared__ float sMu[16], sRs[16];

  const int lane = threadIdx.x;
  const int lo = lane & 15, hi = lane >> 4;
  const int tile = blockIdx.x;

  const float* xs = x_val + (size_t)tile * 16 * 128;
  const float* rsrc = r_val + (size_t)tile * 16 * 128;

  // async-stage both tiles, then form h = x + r in LDS
  stage_tile_async(sX, xs, lane);
  stage_tile_async(sR, rsrc, lane);
  wait_async0();
  __syncthreads();
  for (int i = lane * 4; i < 16 * 128; i += 32 * 4) {
    float4 a = *(float4*)(sX + i);
    const float4 b = *(const float4*)(sR + i);
    a.x += b.x; a.y += b.y; a.z += b.z; a.w += b.w;
    *(float4*)(sX + i) = a;
  }
  __syncthreads();

  for (int nt = 0; nt < 8; ++nt) {
    v8f acc = gemm_tile(sX, W3, 128, 128, nt, lo, hi);
    const float bias = b3[nt * 16 + lo];
    for (int r = 0; r < 8; ++r) {
      float v = acc[r] + bias;
      v = v > 0.0f ? v : 0.0f;
      sT[(r + 8 * hi) * 128 + nt * 16 + lo] = v;
    }
  }
  __syncthreads();

  float rs[8], rq[8];
  for (int r = 0; r < 8; ++r) { rs[r] = 0.0f; rq[r] = 0.0f; }
  for (int nt = 0; nt < 8; ++nt) {
    v8f acc = gemm_tile(sT, W4, 128, 128, nt, lo, hi);
    for (int r = 0; r < 8; ++r) {
      const float v = acc[r];
      sO[(r + 8 * hi) * 128 + nt * 16 + lo] = v;
      rs[r] += v;
      rq[r] += v * v;
    }
  }
  for (int r = 0; r < 8; ++r) {
    float s = rs[r], q = rq[r];
    for (int off = 8; off > 0; off >>= 1) {
      s += __shfl_xor(s, off, 16);
      q += __shfl_xor(q, off, 16);
    }
    if (lo == 0) {
      const int row = r + 8 * hi;
      const float mu = s * (1.0f / 128.0f);
      const float var = q * (1.0f / 128.0f) - mu * mu;
      sMu[row] = mu;
      sRs[row] = rsqrtf(var + 1e-5f);
    }
  }
  __syncthreads();

  float* dst = out + (size_t)tile * 16 * 128;
  for (int i = lane; i < 16 * 128 / 4; i += 32) {
    const int row = i >> 5;            // 32 float4 per row
    const int c4 = (i & 31) * 4;
    float4 v = *(float4*)(sO + row * 128 + c4);
    const float mu = sMu[row], rstd = sRs[row];
    const float4 g = *(const float4*)(g2 + c4);
    const float4 bb = *(const float4*)(bt2 + c4);
    const float4 xv = *(const float4*)(xs + row * 128 + c4);
    v.x = (v.x - mu) * rstd * g.x + bb.x + xv.x;
    v.y = (v.y - mu) * rstd * g.y + bb.y + xv.y;
    v.z = (v.z - mu) * rstd * g.z + bb.z + xv.z;
    v.w = (v.w - mu) * rstd * g.w + bb.w + xv.w;
    *(float4*)(dst + row * 128 + c4) = v;
  }
}

// ---------------------------------------------------------------------------
extern "C" void kernel_launch(void* const* d_in, const int* in_sizes, int n_in,
                              void* d_out, int out_size, void* d_ws, size_t ws_size,
                              hipStream_t stream) {
  const float* x_val = (const float*)d_in[0];
  const float* r_cst = (const float*)d_in[1];
  const int*   edges = (const int*)d_in[2];   // [2, E] flat
  const int*   LE    = (const int*)d_in[3];
  const int*   PE    = (const int*)d_in[4];
  // d_in[5] = num_cst, d_in[6] = num_val (scalars; derive from sizes instead)
  const float* W1  = (const float*)d_in[7];
  const float* b1  = (const float*)d_in[8];
  const float* W2  = (const float*)d_in[9];
  const float* g1  = (const float*)d_in[10];
  const float* bt1 = (const float*)d_in[11];
  const float* W3  = (const float*)d_in[12];
  const float* b3  = (const float*)d_in[13];
  const float* W4  = (const float*)d_in[14];
  const float* g2  = (const float*)d_in[15];
  const float* bt2 = (const float*)d_in[16];

  const int NC = in_sizes[1] / HDIM;
  const int NV = in_sizes[0] / HDIM;
  const int E  = in_sizes[3];

  float* m_ws = (float*)d_ws;        // NC * 512 floats (~205 MB)
  float* out  = (float*)d_out;       // NV * 128 floats; doubles as r_val accumulator

  // 1) constraint-side MLP + LayerNorm -> messages m
  cst_send_kernel<<<NC / 16, 32, 0, stream>>>(r_cst, W1, b1, W2, g1, bt1, m_ws);

  // 2) zero the segment-sum accumulator (d_out)
  hipMemsetAsync(out, 0, (size_t)NV * HDIM * sizeof(float), stream);

  // 3) gather messages per edge, scatter-add into r_val
  const int eblocks = (E + 63) / 64;
  edge_scatter_kernel<<<eblocks, 256, 0, stream>>>(m_ws, edges, edges + E, LE, PE, out, E);

  // 4) value-side MLP + LayerNorm + residual (in place over d_out)
  val_rec_kernel<<<NV / 16, 32, 0, stream>>>(x_val, out, W3, b3, W4, g2, bt2, out);
}